// GINLayer_36335423324483
// MI455X (gfx1250) — compile-verified
//
#include <hip/hip_runtime.h>

// GIN layer for MI455X (gfx1250, wave32):
//   agg = (1+eps)*x + scatter_add(x[col] -> row)
//   h1  = relu(BN1(agg @ W1^T + b1))      (BN stats over 50000 rows)
//   out = relu(BN2(h1  @ W2^T + b2))
//
// GEMMs use V_WMMA_F32_16X16X4_F32 (full fp32 precision, matches reference).
// Everything is HBM-bound; the 410MB edge-gather dominates (~18us @ 23.3TB/s).

typedef float v2f __attribute__((ext_vector_type(2)));
typedef float v8f __attribute__((ext_vector_type(8)));

#define DFEAT 128

// ---------------------------------------------------------------- init: agg = (1+eps)*x
__global__ void gin_init_agg(const float* __restrict__ x,
                             const float* __restrict__ eps,
                             float* __restrict__ agg, int n4) {
  int i = blockIdx.x * blockDim.x + threadIdx.x;
  if (i < n4) {
    const float s = 1.0f + eps[0];
    float4 v = reinterpret_cast<const float4*>(x)[i];
    v.x *= s; v.y *= s; v.z *= s; v.w *= s;
    reinterpret_cast<float4*>(agg)[i] = v;
  }
}

// ---------------------------------------------------------------- edge scatter-add
// One wave per edge; each lane moves 16 bytes (float4) -> 4 f32 atomics.
__global__ void gin_scatter_add(const float* __restrict__ x,
                                const int* __restrict__ row,
                                const int* __restrict__ col,
                                float* __restrict__ agg, int nedges) {
  int t = blockIdx.x * blockDim.x + threadIdx.x;
  int e = t >> 5;
  int lane = t & 31;
  if (e < nedges) {
    int r = row[e];
    int c = col[e];
    float4 v = reinterpret_cast<const float4*>(x + (size_t)c * DFEAT)[lane];
    float* dst = agg + (size_t)r * DFEAT + lane * 4;
    unsafeAtomicAdd(dst + 0, v.x);   // global_atomic_add_f32
    unsafeAtomicAdd(dst + 1, v.y);
    unsafeAtomicAdd(dst + 2, v.z);
    unsafeAtomicAdd(dst + 3, v.w);
  }
}

// ---------------------------------------------------------------- WMMA GEMM (+fused BN/ReLU on A)
// Y[n][o] = sum_k T(A[n][k]) * W[o][k] + bias[o]
//   T = identity          (stage 1)
//   T = relu(a*scale+shift)  per-feature-k (stage 2: applies BN1+ReLU to A on load)
// Block: 256 threads = 8 waves; wave w owns output cols [16w,16w+16); grid.x = N/16 row tiles.
// fp32 WMMA A(16x4) layout: lanes 0-15 -> K=k0,k0+1 ; lanes 16-31 -> K=k0+2,k0+3.
// B(4x16)=W^T mirrors it: per-lane float2 from W[col0+m][k0h] (row-major, contiguous).
template <bool FUSE_BN>
__global__ void gin_gemm_wmma(const float* __restrict__ A,
                              const float* __restrict__ W,
                              const float* __restrict__ bias,
                              const float* __restrict__ scale,
                              const float* __restrict__ shift,
                              float* __restrict__ Y) {
  const int row0 = blockIdx.x * 16;
  const int wave = threadIdx.x >> 5;          // 0..7
  const int lane = threadIdx.x & 31;
  const int col0 = wave * 16;
  const int m    = lane & 15;
  const int hi   = lane >> 4;                 // 0: K lo pair, 1: K hi pair

  const float* __restrict__ Arow = A + (size_t)(row0 + m) * DFEAT;
  const float* __restrict__ Wrow = W + (size_t)(col0 + m) * DFEAT;

  v8f acc = {};
#pragma unroll
  for (int kk = 0; kk < DFEAT / 4; ++kk) {
    const int k = kk * 4 + hi * 2;
    v2f a = *reinterpret_cast<const v2f*>(Arow + k);
    if (FUSE_BN) {
      a.x = fmaxf(0.0f, a.x * scale[k]     + shift[k]);
      a.y = fmaxf(0.0f, a.y * scale[k + 1] + shift[k + 1]);
    }
    v2f b = *reinterpret_cast<const v2f*>(Wrow + k);
    acc = __builtin_amdgcn_wmma_f32_16x16x4_f32(
        /*neg_a=*/false, a, /*neg_b=*/false, b,
        /*c_mod=*/(short)0, acc, /*reuse_a=*/false, /*reuse_b=*/false);
  }

  const float bc = bias[col0 + m];
#pragma unroll
  for (int r = 0; r < 8; ++r) {                 // D VGPR r -> row r + 8*hi, col m
    Y[(size_t)(row0 + r + hi * 8) * DFEAT + col0 + m] = acc[r] + bc;
  }
}

// ---------------------------------------------------------------- column sum / sumsq
// 128 threads = one column each; block covers a strip of rows; block-local
// accumulation in registers, then 2 atomics per column per block (low contention).
__global__ void gin_col_stats(const float* __restrict__ Y,
                              float* __restrict__ stats,  // [0..127]=sum [128..255]=sumsq
                              int n, int rows_per_block) {
  const int c  = threadIdx.x;
  const int r0 = blockIdx.x * rows_per_block;
  const int r1 = min(n, r0 + rows_per_block);
  float s = 0.0f, s2 = 0.0f;
  for (int r = r0; r < r1; ++r) {
    float v = Y[(size_t)r * DFEAT + c];
    s += v;
    s2 = fmaf(v, v, s2);
  }
  unsafeAtomicAdd(&stats[c], s);
  unsafeAtomicAdd(&stats[DFEAT + c], s2);
}

// ---------------------------------------------------------------- BN finalize: scale/shift
__global__ void gin_bn_finalize(const float* __restrict__ stats,
                                const float* __restrict__ gamma,
                                const float* __restrict__ beta,
                                float* __restrict__ ss,   // [0..127]=scale [128..255]=shift
                                float inv_n) {
  const int c = threadIdx.x;
  if (c < DFEAT) {
    const float mu  = stats[c] * inv_n;
    const float var = stats[DFEAT + c] * inv_n - mu * mu;   // biased (ddof=0)
    const float sc  = gamma[c] * rsqrtf(var + 1e-5f);
    ss[c]         = sc;
    ss[DFEAT + c] = beta[c] - mu * sc;
  }
}

// ---------------------------------------------------------------- final BN2 + ReLU
__global__ void gin_bn_relu_out(const float* __restrict__ Y,
                                const float* __restrict__ ss,
                                float* __restrict__ out, int total) {
  int i = blockIdx.x * blockDim.x + threadIdx.x;
  if (i < total) {
    const int c = i & (DFEAT - 1);
    out[i] = fmaxf(0.0f, Y[i] * ss[c] + ss[DFEAT + c]);
  }
}

// ---------------------------------------------------------------- host launcher
extern "C" void kernel_launch(void* const* d_in, const int* in_sizes, int n_in,
                              void* d_out, int out_size, void* d_ws, size_t ws_size,
                              hipStream_t stream) {
  const float* x    = (const float*)d_in[0];
  const int*   ei   = (const int*)  d_in[1];   // [2,E] flat: row then col
  const float* eps  = (const float*)d_in[2];
  const float* W1   = (const float*)d_in[3];
  const float* b1   = (const float*)d_in[4];
  const float* g1   = (const float*)d_in[5];
  const float* be1  = (const float*)d_in[6];
  const float* W2   = (const float*)d_in[7];
  const float* b2   = (const float*)d_in[8];
  const float* g2   = (const float*)d_in[9];
  const float* be2  = (const float*)d_in[10];

  const int n  = in_sizes[0] / DFEAT;    // 50000 (multiple of 16)
  const int ne = in_sizes[1] / 2;        // 800000

  // workspace layout (floats)
  float* agg    = (float*)d_ws;
  float* Y1     = agg + (size_t)n * DFEAT;
  float* Y2     = Y1  + (size_t)n * DFEAT;
  float* stats1 = Y2  + (size_t)n * DFEAT;   // 256
  float* stats2 = stats1 + 2 * DFEAT;        // 256
  float* ss1    = stats2 + 2 * DFEAT;        // 256
  float* ss2    = ss1    + 2 * DFEAT;        // 256

  hipMemsetAsync(stats1, 0, 4 * DFEAT * sizeof(float), stream);  // zero stats1+stats2

  // agg = (1+eps)*x
  const int n4 = n * DFEAT / 4;
  gin_init_agg<<<(n4 + 255) / 256, 256, 0, stream>>>(x, eps, agg, n4);

  // agg[row] += x[col] over edges (one wave per edge)
  const long tthreads = (long)ne * 32;
  gin_scatter_add<<<(int)((tthreads + 255) / 256), 256, 0, stream>>>(
      x, ei, ei + ne, agg, ne);

  // Y1 = agg @ W1^T + b1
  gin_gemm_wmma<false><<<n / 16, 256, 0, stream>>>(agg, W1, b1, nullptr, nullptr, Y1);

  // BN1 stats -> scale/shift
  gin_col_stats<<<(n + 255) / 256, DFEAT, 0, stream>>>(Y1, stats1, n, 256);
  gin_bn_finalize<<<1, DFEAT, 0, stream>>>(stats1, g1, be1, ss1, 1.0f / (float)n);

  // Y2 = relu(BN1(Y1)) @ W2^T + b2   (BN1+ReLU fused into A-load)
  gin_gemm_wmma<true><<<n / 16, 256, 0, stream>>>(Y1, W2, b2, ss1, ss1 + DFEAT, Y2);

  // BN2 stats -> scale/shift
  gin_col_stats<<<(n + 255) / 256, DFEAT, 0, stream>>>(Y2, stats2, n, 256);
  gin_bn_finalize<<<1, DFEAT, 0, stream>>>(stats2, g2, be2, ss2, 1.0f / (float)n);

  // out = relu(BN2(Y2))
  gin_bn_relu_out<<<(n * DFEAT + 255) / 256, 256, 0, stream>>>(Y2, ss2, (float*)d_out, n * DFEAT);
}